// SGC_3135326126431
// MI455X (gfx1250) — compile-verified
//
#include <hip/hip_runtime.h>

// Problem constants (match reference)
#define N_NODES 100000
#define E_EDGES 1600000
#define NFEAT   128
#define NOUT    64
#define P_DROP  0.1f

typedef __attribute__((ext_vector_type(2))) float v2f;
typedef __attribute__((ext_vector_type(8))) float v8f;

// ---------------------------------------------------------------------------
// Kernel 1: zero the aggregation workspace (d_ws is poisoned each run)
// ---------------------------------------------------------------------------
__global__ void zero_agg(float4* __restrict__ agg4, int n4) {
    int stride = gridDim.x * blockDim.x;
    for (int i = blockIdx.x * blockDim.x + threadIdx.x; i < n4; i += stride) {
        agg4[i] = make_float4(0.f, 0.f, 0.f, 0.f);
    }
}

// ---------------------------------------------------------------------------
// Kernel 2: COO SpMM via L2-resident atomic scatter.
// One wave32 per edge: lane L loads x[col][4L..4L+3] (float4, coalesced 512B
// row) and atomically adds val*x into agg[row]. x and agg both fit in the
// 192MB L2, so these atomics are serviced at L2 bandwidth.
// ---------------------------------------------------------------------------
__global__ void spmm_scatter(const float* __restrict__ x,
                             const int*   __restrict__ rows,
                             const int*   __restrict__ cols,
                             const float* __restrict__ vals,
                             float*       __restrict__ agg) {
    const int lane   = threadIdx.x & 31;
    const int wid    = (blockIdx.x * blockDim.x + threadIdx.x) >> 5;
    const int nwaves = (gridDim.x * blockDim.x) >> 5;

    for (int e = wid; e < E_EDGES; e += nwaves) {
        const int   r = rows[e];
        const int   c = cols[e];
        const float v = vals[e];
        float4 xv = ((const float4*)(x + (size_t)c * NFEAT))[lane];
        float* dst = agg + (size_t)r * NFEAT + lane * 4;
        atomicAdd(dst + 0, xv.x * v);
        atomicAdd(dst + 1, xv.y * v);
        atomicAdd(dst + 2, xv.z * v);
        atomicAdd(dst + 3, xv.w * v);
    }
}

// ---------------------------------------------------------------------------
// Kernel 3: fused dropout + dense GEMM using V_WMMA_F32_16X16X4_F32.
// out[m][n] = sum_k (dropout(agg[m][k])) * W[n][k] + b[n]
// One wave per 16-row tile; 4 column tiles (NOUT=64) accumulated in parallel.
//
// A-fragment layout (16x4 f32): lanes 0-15 -> M=lane, K={0,1} in {v0,v1};
//                               lanes 16-31 -> M=lane-16, K={2,3}.
// B-fragment layout (4x16 f32, mirrored): lanes 0-15 -> N=lane, K={0,1};
//                               lanes 16-31 -> N=lane-16, K={2,3}.
// C/D layout (16x16 f32): VGPR i, lane L<16 -> (M=i,   N=L),
//                                  lane L>=16 -> (M=i+8, N=L-16).
// => each lane owns one N column: bias is a single scalar add per lane.
// ---------------------------------------------------------------------------
__global__ void __launch_bounds__(128)
gemm_wmma(const float* __restrict__ agg,
          const float* __restrict__ du,
          const float* __restrict__ W,
          const float* __restrict__ bias,
          float*       __restrict__ out) {
    const int lane     = threadIdx.x & 31;
    const int wave     = threadIdx.x >> 5;
    const int row_tile = blockIdx.x * 4 + wave;
    if (row_tile * 16 >= N_NODES) return;   // wave-uniform: EXEC stays all-1s

    const int   row_base = row_tile * 16;
    const int   half     = lane >> 4;       // 0: K={0,1}, 1: K={2,3}
    const int   l16      = lane & 15;
    const float inv_keep = 1.0f / (1.0f - P_DROP);

    v8f acc0 = {}, acc1 = {}, acc2 = {}, acc3 = {};

    const float* arow = agg + (size_t)(row_base + l16) * NFEAT + half * 2;
    const float* urow = du  + (size_t)(row_base + l16) * NFEAT + half * 2;
    const float* w0   = W + (size_t)( 0 + l16) * NFEAT + half * 2;
    const float* w1   = W + (size_t)(16 + l16) * NFEAT + half * 2;
    const float* w2   = W + (size_t)(32 + l16) * NFEAT + half * 2;
    const float* w3   = W + (size_t)(48 + l16) * NFEAT + half * 2;

#pragma unroll
    for (int k0 = 0; k0 < NFEAT; k0 += 4) {
        float2 av = *(const float2*)(arow + k0);
        float2 uv = *(const float2*)(urow + k0);
        v2f a;
        a.x = (uv.x >= P_DROP) ? av.x * inv_keep : 0.0f;
        a.y = (uv.y >= P_DROP) ? av.y * inv_keep : 0.0f;

        float2 wv0 = *(const float2*)(w0 + k0);
        float2 wv1 = *(const float2*)(w1 + k0);
        float2 wv2 = *(const float2*)(w2 + k0);
        float2 wv3 = *(const float2*)(w3 + k0);
        v2f b0; b0.x = wv0.x; b0.y = wv0.y;
        v2f b1; b1.x = wv1.x; b1.y = wv1.y;
        v2f b2; b2.x = wv2.x; b2.y = wv2.y;
        v2f b3; b3.x = wv3.x; b3.y = wv3.y;

        // 8-arg form: (neg_a, A, neg_b, B, c_mod, C, reuse_a, reuse_b)
        acc0 = __builtin_amdgcn_wmma_f32_16x16x4_f32(false, a, false, b0,
                                                     (short)0, acc0, false, false);
        acc1 = __builtin_amdgcn_wmma_f32_16x16x4_f32(false, a, false, b1,
                                                     (short)0, acc1, false, false);
        acc2 = __builtin_amdgcn_wmma_f32_16x16x4_f32(false, a, false, b2,
                                                     (short)0, acc2, false, false);
        acc3 = __builtin_amdgcn_wmma_f32_16x16x4_f32(false, a, false, b3,
                                                     (short)0, acc3, false, false);
    }

    // Epilogue: bias + store. Lane owns column n = t*16 + l16 across all VGPRs.
    const int m_off = half * 8;
#pragma unroll
    for (int t = 0; t < 4; ++t) {
        v8f acc = (t == 0) ? acc0 : (t == 1) ? acc1 : (t == 2) ? acc2 : acc3;
        const int   n  = t * 16 + l16;
        const float bv = bias[n];
#pragma unroll
        for (int i = 0; i < 8; ++i) {
            const int m = row_base + m_off + i;
            out[(size_t)m * NOUT + n] = acc[i] + bv;
        }
    }
}

// ---------------------------------------------------------------------------
extern "C" void kernel_launch(void* const* d_in, const int* in_sizes, int n_in,
                              void* d_out, int out_size, void* d_ws, size_t ws_size,
                              hipStream_t stream) {
    const float* x    = (const float*)d_in[0];
    const int*   rows = (const int*)  d_in[1];
    const int*   cols = (const int*)  d_in[2];
    const float* vals = (const float*)d_in[3];
    const float* du   = (const float*)d_in[4];
    const float* W    = (const float*)d_in[5];
    const float* bias = (const float*)d_in[6];
    float*       out  = (float*)d_out;
    float*       agg  = (float*)d_ws;     // N_NODES * NFEAT fp32 = 51.2 MB

    // 1) zero the segment-sum accumulator
    zero_agg<<<2048, 256, 0, stream>>>((float4*)agg, N_NODES * NFEAT / 4);

    // 2) edge scatter (wave-per-edge, 8192 persistent waves, grid-stride)
    spmm_scatter<<<1024, 256, 0, stream>>>(x, rows, cols, vals, agg);

    // 3) fused dropout + WMMA GEMM + bias (6250 row tiles, 4 waves/block)
    gemm_wmma<<<(N_NODES / 16 + 3) / 4, 128, 0, stream>>>(agg, du, W, bias, out);
}